// MIX_69698729280145
// MI455X (gfx1250) — compile-verified
//
#include <hip/hip_runtime.h>
#include <hip/hip_bf16.h>
#include <stdint.h>

// ---------------------------------------------------------------------------
// Streaming mixture-of-activations kernel for MI455X (gfx1250).
// ~128MB HBM traffic (~5.5us roofline @ 23.3TB/s) but VALU-limited, so:
//  - s tiles staged into LDS via Tensor Data Mover (async, double buffered)
//  - per-neuron 4x4 weights/bias pre-scaled by log2(e) once per thread so the
//    4-way softmax needs only raw v_exp_f32 (exp2) ops, no stabilization.
// ---------------------------------------------------------------------------

typedef __attribute__((ext_vector_type(4))) unsigned int uint32x4;
typedef __attribute__((ext_vector_type(8))) int          int32x8;
typedef __attribute__((ext_vector_type(4))) int          int32x4;

#define TN    256   // neurons per block (== blockDim.x)
#define CHUNK 16    // batch rows per TDM tile
#define NCHNK 4     // tiles per block
#define ROWS  (CHUNK * NCHNK)

#define LOG2E 1.4426950408889634f

// Issue a 2D TDM load: tile of CHUNK rows x TN floats from global into LDS.
// Must be executed by exactly one wave of the workgroup.
__device__ __forceinline__ void tdm_load_tile(const float* gptr,
                                              unsigned int lds_off,
                                              int row_stride_elems) {
  unsigned long long ga = (unsigned long long)(uintptr_t)gptr;
  const unsigned int dim0 = (unsigned int)row_stride_elems; // tensor dim0 (>= tile_dim0)
  const unsigned int dim1 = 1u << 20;                       // large: no OOB clipping

  uint32x4 g0;
  g0[0] = 1u;                                   // count=1 (valid user descriptor)
  g0[1] = lds_off;                              // lds_addr (bytes)
  g0[2] = (unsigned int)ga;                     // global_addr[31:0]
  g0[3] = (unsigned int)((ga >> 32) & 0x1FFFFFFull) | (2u << 30); // addr[56:32] | type=2

  int32x8 g1;
  g1[0] = (int)(2u << 16);                                   // data_size = 4 bytes
  g1[1] = (int)((dim0 & 0xFFFFu) << 16);                     // tensor_dim0[15:0]
  g1[2] = (int)(((dim0 >> 16) & 0xFFFFu) | ((dim1 & 0xFFFFu) << 16));
  g1[3] = (int)(((dim1 >> 16) & 0xFFFFu) | ((unsigned)TN << 16)); // tile_dim0 = 256
  g1[4] = (int)(CHUNK & 0xFFFF);                             // tile_dim1 = 16, tile_dim2 = 0
  g1[5] = (int)(unsigned)row_stride_elems;                   // tensor_dim0_stride[31:0]
  g1[6] = 0;                                                 // stride0 hi | stride1 lo
  g1[7] = 0;

  int32x4 g2 = {0, 0, 0, 0};              // 2D tensor: groups 2/3 unused
  int32x4 g3 = {0, 0, 0, 0};
  int32x8 g4 = {0, 0, 0, 0, 0, 0, 0, 0};  // extra group on clang-23 toolchain (unused)

  __builtin_amdgcn_tensor_load_to_lds(g0, g1, g2, g3, g4, 0);
}

// w/bb are pre-scaled by LOG2E: logits computed directly in log2 domain.
__device__ __forceinline__ float mix_elem(float x, const float4* w, float4 bb) {
  // experts (base-2 forms: one v_exp + one v_rcp each)
  float a0 = fmaxf(x, 0.0f);                                              // relu
  float a1 = __builtin_amdgcn_rcpf(1.0f + __builtin_amdgcn_exp2f(-LOG2E * x));          // sigmoid
  float a2 = 2.0f * __builtin_amdgcn_rcpf(1.0f + __builtin_amdgcn_exp2f(-2.0f * LOG2E * x)) - 1.0f; // tanh
  float a3 = x;
  // per-neuron 4x4 linear in log2 domain: l_j = log2e*(b_j + sum_k a_k W[k][j])
  float l0 = bb.x + a0 * w[0].x + a1 * w[1].x + a2 * w[2].x + a3 * w[3].x;
  float l1 = bb.y + a0 * w[0].y + a1 * w[1].y + a2 * w[2].y + a3 * w[3].y;
  float l2 = bb.z + a0 * w[0].z + a1 * w[1].z + a2 * w[2].z + a3 * w[3].z;
  float l3 = bb.w + a0 * w[0].w + a1 * w[1].w + a2 * w[2].w + a3 * w[3].w;
  // unnormalized softmax weights (|l| small enough that exp2 can't overflow)
  float p0 = __builtin_amdgcn_exp2f(l0), p1 = __builtin_amdgcn_exp2f(l1);
  float p2 = __builtin_amdgcn_exp2f(l2), p3 = __builtin_amdgcn_exp2f(l3);
  float den = (p0 + p1) + (p2 + p3);
  float num = p0 * a0 + p1 * a1 + p2 * a2 + p3 * a3;
  return num * __builtin_amdgcn_rcpf(den);
}

__global__ __launch_bounds__(TN) void moa_mix_kernel(const float* __restrict__ s,
                                                     const float* __restrict__ W,
                                                     const float* __restrict__ bias,
                                                     float* __restrict__ out,
                                                     int N) {
  __shared__ float smem[2][CHUNK][TN];   // 32 KB, double buffered s tiles

  const int tid = threadIdx.x;
  const int n   = blockIdx.x * TN + tid;
  const int b0  = blockIdx.y * ROWS;

  // Per-neuron params -> registers, pre-scaled by log2(e) (amortized over ROWS rows).
  float4 w[4];
  const float4* Wv = (const float4*)(W + (size_t)n * 16);
  #pragma unroll
  for (int k = 0; k < 4; ++k) {
    float4 t = Wv[k];
    w[k] = make_float4(t.x * LOG2E, t.y * LOG2E, t.z * LOG2E, t.w * LOG2E);
  }
  float4 bt = *(const float4*)(bias + (size_t)n * 4);
  float4 bb = make_float4(bt.x * LOG2E, bt.y * LOG2E, bt.z * LOG2E, bt.w * LOG2E);

  const unsigned int lds_base = (unsigned int)(uintptr_t)&smem[0][0][0];
  const float* tile0 = s + (size_t)b0 * N + blockIdx.x * TN;

  if (tid < 32) {  // wave 0 drives the TDM pipeline
    tdm_load_tile(tile0, lds_base, N);
  }

  for (int c = 0; c < NCHNK; ++c) {
    const int p = c & 1;
    if (tid < 32) {
      if (c + 1 < NCHNK) {
        // prefetch next tile into the other buffer (safe: all waves finished
        // reading it at the trailing barrier of iteration c-1)
        tdm_load_tile(tile0 + (size_t)(c + 1) * CHUNK * N,
                      lds_base + (unsigned int)((c + 1) & 1) * CHUNK * TN * 4, N);
        __builtin_amdgcn_s_wait_tensorcnt(1);  // tile c has landed
      } else {
        __builtin_amdgcn_s_wait_tensorcnt(0);  // last tile has landed
      }
    }
    __syncthreads();  // release compute on tile c

    #pragma unroll
    for (int r = 0; r < CHUNK; ++r) {
      float x = smem[p][r][tid];
      out[(size_t)(b0 + c * CHUNK + r) * N + n] = mix_elem(x, w, bb);
    }

    __syncthreads();  // everyone done reading buffer p before it is re-filled
  }
}

extern "C" void kernel_launch(void* const* d_in, const int* in_sizes, int n_in,
                              void* d_out, int out_size, void* d_ws, size_t ws_size,
                              hipStream_t stream) {
  const float* s    = (const float*)d_in[0];  // [B, N]
  const float* W    = (const float*)d_in[1];  // [N, 4, 4]
  const float* bias = (const float*)d_in[2];  // [N, 4]
  float* out        = (float*)d_out;          // [B, N]

  const int N = in_sizes[2] / 4;              // 2048
  const int B = in_sizes[0] / N;              // 8192

  dim3 block(TN);
  dim3 grid(N / TN, B / ROWS);
  moa_mix_kernel<<<grid, block, 0, stream>>>(s, W, bias, out, N);
}